// SynthesisPrologue_73830487818665
// MI455X (gfx1250) — compile-verified
//
#include <hip/hip_runtime.h>

// ---------------------------------------------------------------------------
// SynthesisPrologue for MI455X (gfx1250, wave32).
// Main conv = per-batch GEMM (M=65536,N=128,K=1152) on V_WMMA_F32_16X16X4_F32.
// Memory-bound (gather is L2-resident); f32 kept end-to-end for exactness.
// ---------------------------------------------------------------------------

#define F_FACES 65536
#define CCH     128
#define WDIM    512
#define KK      1152            // 9 * 128 contraction length
#define KKP     576             // KK / 2 (kk-pairs)
#define TILE_F  32              // faces per workgroup in main conv
#define LDS_STRIDE 1156         // 1152 + 4 floats pad (bank stagger)
#define SQRT2   1.41421356237309515f

typedef float v2f __attribute__((ext_vector_type(2)));
typedef float v8f __attribute__((ext_vector_type(8)));

// workspace layout (in floats)
#define WS_WPACK   0                       // [B][KKP][CCH][2] : B*576*128*2
#define WS_W2MOD   (2*KKP*CCH*2)           // [B][3][CCH]
#define WS_STYLES  (WS_W2MOD + 2*3*CCH)    // [B][CCH]
#define WS_STYLES2 (WS_STYLES + 2*CCH)     // [B][CCH]
#define WS_DEMOD   (WS_STYLES2 + 2*CCH)    // [B][CCH]

// ---------------------------------------------------------------------------
// Setup 1: styles / styles2 / demod factors (tiny, one workgroup)
// ---------------------------------------------------------------------------
__global__ void k_setup1(const float* __restrict__ wsv,   // (B,2,512)
                         const float* __restrict__ a1W, const float* __restrict__ a1b,
                         const float* __restrict__ a2W, const float* __restrict__ a2b,
                         const float* __restrict__ w1,    // (128,128,1,9)
                         float* __restrict__ wsp, int B)
{
    const int t  = threadIdx.x;
    const int nb = B * CCH;                 // B <= 2 here
    __shared__ float s_styles[2 * CCH];
    const float gain = 0.044194173824159216f;  // 1/sqrt(512)
    const float wg   = 0.08838834764831845f;   // 1/sqrt(128)

    if (t < nb) {
        const int b = t / CCH, i = t % CCH;
        const float* wv0 = wsv + (size_t)b * 2 * WDIM;       // ws[b,0,:]
        const float* wv1 = wv0 + WDIM;                       // ws[b,1,:]
        const float* r1  = a1W + (size_t)i * WDIM;
        const float* r2  = a2W + (size_t)i * WDIM;
        float s1 = 0.f, s2 = 0.f;
        for (int j = 0; j < WDIM; ++j) { s1 += wv0[j] * r1[j]; s2 += wv1[j] * r2[j]; }
        s1 = s1 * gain + a1b[i];
        s2 = (s2 * gain + a2b[i]) * wg;
        s_styles[t]            = s1;
        wsp[WS_STYLES  + t]    = s1;
        wsp[WS_STYLES2 + t]    = s2;
    }
    __syncthreads();
    if (t < nb) {
        const int b = t / CCH, o = t % CCH;
        const float* wrow = w1 + (size_t)o * CCH * 9;
        float acc = 0.f;
        for (int i = 0; i < CCH; ++i) {
            const float st = s_styles[b * CCH + i];
            #pragma unroll
            for (int k = 0; k < 9; ++k) {
                const float v = wrow[i * 9 + k] * st;
                acc += v * v;
            }
        }
        wsp[WS_DEMOD + t] = 1.0f / sqrtf(acc + 1e-8f);
    }
}

// ---------------------------------------------------------------------------
// Setup 2: modulated+demodulated weight pack for WMMA B-fragments
//   Wpack[b][kkp][o][j] = W1[o][i][k] * styles[b][i] * d[b][o],  kk=2*kkp+j,
//   k = kk/128, i = kk%128  (so a lane's (b0,b1) K-pair is one 8B load)
// Also w2mod[b][o][i] = W2[o][i] * styles2[b][i].
// ---------------------------------------------------------------------------
__global__ void k_setup2(const float* __restrict__ w1, const float* __restrict__ w2,
                         float* __restrict__ wsp, int B)
{
    const float* styles  = wsp + WS_STYLES;
    const float* styles2 = wsp + WS_STYLES2;
    const float* dmod    = wsp + WS_DEMOD;
    float* wpack = wsp + WS_WPACK;
    float* w2mod = wsp + WS_W2MOD;
    const int total1 = B * KKP * CCH * 2;
    const int total2 = B * 3 * CCH;
    for (int e = blockIdx.x * blockDim.x + threadIdx.x; e < total1 + total2;
         e += gridDim.x * blockDim.x) {
        if (e < total1) {
            const int b   = e / (KKP * CCH * 2);
            const int r   = e % (KKP * CCH * 2);
            const int kkp = r >> 8;          // / 256
            const int q   = r & 255;
            const int o   = q >> 1, j = q & 1;
            const int kk  = kkp * 2 + j;
            const int k   = kk >> 7, i = kk & 127;
            wpack[e] = w1[((size_t)o * CCH + i) * 9 + k]
                     * styles[b * CCH + i] * dmod[b * CCH + o];
        } else {
            const int e2 = e - total1;
            const int b  = e2 / (3 * CCH);
            const int r  = e2 % (3 * CCH);
            const int o  = r / CCH, i = r % CCH;
            w2mod[e2] = w2[o * CCH + i] * styles2[b * CCH + i];
        }
    }
}

// ---------------------------------------------------------------------------
// Main modulated face conv: gather tile -> LDS once, then WMMA f32 16x16x4.
// Workgroup: 512 threads = 16 waves; tile = 32 faces x 128 outs.
// Wave w: faces (w&1)*16..+15, outs (w>>1)*16..+15; 288 WMMAs over K=1152.
// ---------------------------------------------------------------------------
__global__ __launch_bounds__(512) void k_conv1(
    const long long* __restrict__ neigh,          // (N,9) int64
    const unsigned char* __restrict__ ispad,      // (N,9) bool
    const float* __restrict__ cst,                // (F,128); x row r = cst[r%F]
    const float* __restrict__ wpack,              // WS_WPACK
    const float* __restrict__ noise_c,            // (F,)
    const float* __restrict__ noise_s,            // (1,)
    const float* __restrict__ bias1,              // (128,)
    float* __restrict__ xout,                     // (N,128)
    int Ntot)
{
    extern __shared__ float g_lds[];              // TILE_F * LDS_STRIDE floats
    const int tid   = threadIdx.x;
    const int w     = tid >> 5;                   // wave id (0..15)
    const int lane  = tid & 31;
    const int fbase = blockIdx.x * TILE_F;
    const int b     = fbase / F_FACES;            // batch (F multiple of TILE_F)

    // ---- Phase A: gather 32 faces x 9 neighbors x 128 f32 into LDS --------
    // One wave copies one (face,neighbor) 512B row per iteration.
    #pragma unroll
    for (int it = 0; it < (TILE_F * 9) / 16; ++it) {
        const int r = w + 16 * it;                // 0..287
        const int f = r / 9, k = r % 9;
        const size_t ei = (size_t)(fbase + f) * 9 + k;
        const long long nbi = neigh[ei];
        const bool zero = (ispad[ei] != 0) || (nbi >= (long long)Ntot);
        float4 v = make_float4(0.f, 0.f, 0.f, 0.f);
        if (!zero) {
            const int src = (int)(nbi % F_FACES); // tiled const row
            v = *(const float4*)(cst + (size_t)src * CCH + lane * 4);
        }
        *(float4*)(&g_lds[f * LDS_STRIDE + k * CCH + lane * 4]) = v;
    }
    __syncthreads();

    // ---- Phase B: 16x16 output tile via V_WMMA_F32_16X16X4_F32 ------------
    const int fh   = w & 1;                       // face half
    const int ot   = w >> 1;                      // out tile
    const int row  = lane & 15;
    const int half = lane >> 4;
    const int o    = ot * 16 + row;
    // A fragment: element (M=row, K=2*half + vgpr)
    const float* ap = &g_lds[(fh * 16 + row) * LDS_STRIDE + 2 * half];
    // B fragment: element (K=2*half + vgpr, N=row) -> Wpack[b][2t+half][o][0..1]
    const float* bp = wpack + (size_t)b * (KKP * CCH * 2) + half * (CCH * 2) + o * 2;

    v8f acc = {};
    #pragma unroll 4
    for (int t = 0; t < KK / 4; ++t) {
        const v2f a  = *(const v2f*)(ap + 4 * t);
        const v2f bb = *(const v2f*)(bp + (size_t)t * (2 * CCH * 2));
        acc = __builtin_amdgcn_wmma_f32_16x16x4_f32(
                  false, a, false, bb, (short)0, acc, false, false);
    }

    // ---- Epilogue: +noise, +bias, leaky-relu, *sqrt2, clip ----------------
    const float b1  = bias1[o];
    const float nst = noise_s[0];
    #pragma unroll
    for (int j = 0; j < 8; ++j) {
        const int n = fbase + fh * 16 + j + 8 * half;   // C/D layout: M=j+8*half
        float v = acc[j] + noise_c[n & (F_FACES - 1)] * nst + b1;
        v = (v > 0.f) ? v : 0.2f * v;
        v *= SQRT2;
        v = fminf(fmaxf(v, -256.f), 256.f);
        xout[(size_t)n * CCH + o] = v;
    }
}

// ---------------------------------------------------------------------------
// Second conv: img[n,o] = clip( dot(x[neigh[n,0]], w2mod[b][o]) + bias2[o] )
// ---------------------------------------------------------------------------
__global__ void k_conv2(const long long* __restrict__ neigh,
                        const unsigned char* __restrict__ ispad,
                        const float* __restrict__ xin,     // (N,128) = output 1
                        const float* __restrict__ w2mod,   // [B][3][128]
                        const float* __restrict__ bias2,   // (3,)
                        float* __restrict__ img,           // (N,3)
                        int Ntot, int B)
{
    __shared__ float s_w2[2 * 3 * CCH];
    for (int t = threadIdx.x; t < B * 3 * CCH; t += blockDim.x) s_w2[t] = w2mod[t];
    __syncthreads();

    const int n = blockIdx.x * blockDim.x + threadIdx.x;
    if (n >= Ntot) return;
    const int b = n / F_FACES;
    const size_t ei = (size_t)n * 9;
    const long long nbi = neigh[ei];
    const bool zero = (ispad[ei] != 0) || (nbi >= (long long)Ntot);

    float a0 = 0.f, a1 = 0.f, a2 = 0.f;
    if (!zero) {
        const float4* xr = (const float4*)(xin + (size_t)nbi * CCH);
        const float* w0r = &s_w2[(b * 3 + 0) * CCH];
        const float* w1r = &s_w2[(b * 3 + 1) * CCH];
        const float* w2r = &s_w2[(b * 3 + 2) * CCH];
        #pragma unroll 8
        for (int ii = 0; ii < CCH / 4; ++ii) {
            const float4 xv = xr[ii];
            const int i = ii * 4;
            a0 += xv.x * w0r[i] + xv.y * w0r[i + 1] + xv.z * w0r[i + 2] + xv.w * w0r[i + 3];
            a1 += xv.x * w1r[i] + xv.y * w1r[i + 1] + xv.z * w1r[i + 2] + xv.w * w1r[i + 3];
            a2 += xv.x * w2r[i] + xv.y * w2r[i + 1] + xv.z * w2r[i + 2] + xv.w * w2r[i + 3];
        }
    }
    img[(size_t)n * 3 + 0] = fminf(fmaxf(a0 + bias2[0], -256.f), 256.f);
    img[(size_t)n * 3 + 1] = fminf(fmaxf(a1 + bias2[1], -256.f), 256.f);
    img[(size_t)n * 3 + 2] = fminf(fmaxf(a2 + bias2[2], -256.f), 256.f);
}

// ---------------------------------------------------------------------------
extern "C" void kernel_launch(void* const* d_in, const int* in_sizes, int n_in,
                              void* d_out, int out_size, void* d_ws, size_t ws_size,
                              hipStream_t stream)
{
    const long long*     neigh   = (const long long*)d_in[0];
    const unsigned char* ispad   = (const unsigned char*)d_in[1];
    const float*         wsv     = (const float*)d_in[2];
    const float*         cst     = (const float*)d_in[3];
    const float*         a1W     = (const float*)d_in[4];
    const float*         a1b     = (const float*)d_in[5];
    const float*         w1      = (const float*)d_in[6];
    const float*         noise_c = (const float*)d_in[7];
    const float*         noise_s = (const float*)d_in[8];
    const float*         bias1   = (const float*)d_in[9];
    const float*         a2W     = (const float*)d_in[10];
    const float*         a2b     = (const float*)d_in[11];
    const float*         w2      = (const float*)d_in[12];
    const float*         bias2   = (const float*)d_in[13];

    const int B    = in_sizes[2] / (2 * WDIM);   // 2
    const int Ntot = B * F_FACES;                // 131072
    float* wsp  = (float*)d_ws;
    float* xout = (float*)d_out;                       // first output: (N,128)
    float* img  = xout + (size_t)Ntot * CCH;           // second output: (N,3)

    k_setup1<<<1, 256, 0, stream>>>(wsv, a1W, a1b, a2W, a2b, w1, wsp, B);
    k_setup2<<<256, 256, 0, stream>>>(w1, w2, wsp, B);

    const int nblk = Ntot / TILE_F;                    // 4096
    const size_t lds_bytes = (size_t)TILE_F * LDS_STRIDE * sizeof(float); // ~144.5 KB
    k_conv1<<<nblk, 512, lds_bytes, stream>>>(neigh, ispad, cst, wsp + WS_WPACK,
                                              noise_c, noise_s, bias1, xout, Ntot);

    k_conv2<<<(Ntot + 255) / 256, 256, 0, stream>>>(neigh, ispad, xout,
                                                    wsp + WS_W2MOD, bias2, img,
                                                    Ntot, B);
}